// GNNStack_backbone_1254130450726
// MI455X (gfx1250) — compile-verified
//
#include <hip/hip_runtime.h>

typedef __attribute__((ext_vector_type(2))) float v2f;
typedef __attribute__((ext_vector_type(8))) float v8f;

#define NN 50000
#define NE 800000
#define HD 128

// ---------------------------------------------------------------- utilities
__global__ __launch_bounds__(256) void k_zero(float* __restrict__ p, int n) {
  int i = blockIdx.x * 256 + threadIdx.x;
  if (i < n) p[i] = 0.0f;
}

__global__ __launch_bounds__(256) void k_degree(const int* __restrict__ dst,
                                                float* __restrict__ deg) {
  int e = blockIdx.x * 256 + threadIdx.x;
  if (e < NE) unsafeAtomicAdd(&deg[dst[e]], 1.0f);   // global_atomic_add_f32
}

__global__ __launch_bounds__(256) void k_dinv(float* __restrict__ deg) {
  int i = blockIdx.x * 256 + threadIdx.x;
  if (i < NN) deg[i] = rsqrtf(deg[i] + 1.0f);
}

// ---------------------------------------------------------------- WMMA GEMM
// H[row0:row0+16, col0:col0+16] = X[row0:row0+16, :] @ W[:, col0:col0+16]
// One 16x16 output tile per wave; 8 waves/block cover all 128 columns.
// fp32 WMMA: A is 16x4 (2 f32/lane), B is 4x16 (2 f32/lane), C/D 16x16 (8 f32/lane).
// Lane map (ISA 7.12.2): lanes 0-15 -> K = k0,k0+1 ; lanes 16-31 -> K = k0+2,k0+3.
__global__ __launch_bounds__(256) void k_gemm(const float* __restrict__ X,
                                              const float* __restrict__ W,
                                              float* __restrict__ H) {
  const int wave  = threadIdx.x >> 5;
  const int lane  = threadIdx.x & 31;
  const int laneM = lane & 15;
  const int hi    = lane >> 4;          // 0 or 1 -> K sub-pair select
  const int row0  = blockIdx.x << 4;
  const int col0  = wave << 4;

  const float* __restrict__ xr = X + (size_t)(row0 + laneM) * HD;  // A row
  const float* __restrict__ wc = W + col0 + laneM;                 // B column

  v8f c = {};
#pragma unroll 4
  for (int k0 = 0; k0 < HD; k0 += 4) {
    const int ka = k0 + (hi << 1);
    v2f a, b;
    a.x = xr[ka];
    a.y = xr[ka + 1];
    b.x = wc[(size_t)ka * HD];
    b.y = wc[(size_t)(ka + 1) * HD];
    c = __builtin_amdgcn_wmma_f32_16x16x4_f32(false, a, false, b,
                                              (short)0, c, false, false);
  }

  // D layout: VGPR j -> M = j (lanes 0-15) / j+8 (lanes 16-31), N = laneM
  float* __restrict__ ho = H + col0 + laneM;
#pragma unroll
  for (int j = 0; j < 8; ++j)
    ho[(size_t)(row0 + j + (hi << 3)) * HD] = c[j];
}

// ---------------------------------------------------------------- edge scatter
// One wave32 per edge: lane loads float4 of h[src] (global_load_b128) and does
// 4 hardware fp32 atomics into agg[dst]. src/dst/dinv reads are wave-uniform
// (scalarized by the compiler).
__global__ __launch_bounds__(256) void k_scatter(const int* __restrict__ src,
                                                 const int* __restrict__ dst,
                                                 const float* __restrict__ dinv,
                                                 const float* __restrict__ H,
                                                 float* __restrict__ agg) {
  const int e = (blockIdx.x * 256 + threadIdx.x) >> 5;
  const int lane = threadIdx.x & 31;
  if (e >= NE) return;
  const int s = src[e];
  const int d = dst[e];
  const float nrm = dinv[s] * dinv[d];
  const float4 v = ((const float4*)(H + (size_t)s * HD))[lane];
  float* a = agg + (size_t)d * HD + lane * 4;
  unsafeAtomicAdd(a + 0, v.x * nrm);
  unsafeAtomicAdd(a + 1, v.y * nrm);
  unsafeAtomicAdd(a + 2, v.z * nrm);
  unsafeAtomicAdd(a + 3, v.w * nrm);
}

// ---------------------------------------------------------------- epilogue
// One wave32 per node row: emb = agg + dinv^2 * h + bias ; x = relu(emb) ;
// optional LayerNorm via wave32 shuffle reductions (4 features per lane).
__global__ __launch_bounds__(256) void k_post(const float* __restrict__ agg,
                                              const float* __restrict__ H,
                                              const float* __restrict__ dinv,
                                              const float* __restrict__ bias,
                                              const float* __restrict__ gamma,
                                              const float* __restrict__ beta,
                                              float* __restrict__ xout,
                                              float* __restrict__ embout,
                                              int doLN) {
  const int n = (blockIdx.x * 256 + threadIdx.x) >> 5;
  const int lane = threadIdx.x & 31;
  if (n >= NN) return;
  const float di = dinv[n];
  const float sw = di * di;
  const float4 a = ((const float4*)(agg + (size_t)n * HD))[lane];
  const float4 h = ((const float4*)(H + (size_t)n * HD))[lane];
  const float4 b = ((const float4*)bias)[lane];
  float4 e;
  e.x = fmaf(sw, h.x, a.x) + b.x;
  e.y = fmaf(sw, h.y, a.y) + b.y;
  e.z = fmaf(sw, h.z, a.z) + b.z;
  e.w = fmaf(sw, h.w, a.w) + b.w;
  if (embout) ((float4*)(embout + (size_t)n * HD))[lane] = e;

  float4 r;
  r.x = fmaxf(e.x, 0.f); r.y = fmaxf(e.y, 0.f);
  r.z = fmaxf(e.z, 0.f); r.w = fmaxf(e.w, 0.f);

  if (doLN) {
    float s = r.x + r.y + r.z + r.w;
#pragma unroll
    for (int o = 16; o; o >>= 1) s += __shfl_xor(s, o, 32);
    const float mu = s * (1.0f / HD);
    float4 dlt = { r.x - mu, r.y - mu, r.z - mu, r.w - mu };
    float s2 = dlt.x * dlt.x + dlt.y * dlt.y + dlt.z * dlt.z + dlt.w * dlt.w;
#pragma unroll
    for (int o = 16; o; o >>= 1) s2 += __shfl_xor(s2, o, 32);
    const float rstd = rsqrtf(s2 * (1.0f / HD) + 1e-5f);
    const float4 g  = ((const float4*)gamma)[lane];
    const float4 bt = ((const float4*)beta)[lane];
    r.x = fmaf(dlt.x * rstd, g.x, bt.x);
    r.y = fmaf(dlt.y * rstd, g.y, bt.y);
    r.z = fmaf(dlt.z * rstd, g.z, bt.z);
    r.w = fmaf(dlt.w * rstd, g.w, bt.w);
  }
  ((float4*)(xout + (size_t)n * HD))[lane] = r;
}

// ---------------------------------------------------------------- driver
extern "C" void kernel_launch(void* const* d_in, const int* in_sizes, int n_in,
                              void* d_out, int out_size, void* d_ws, size_t ws_size,
                              hipStream_t stream) {
  // dict order: x, edge_index, Ws[0..2], bs[0..2], gammas[0..2], betas[0..2]
  const float* x0  = (const float*)d_in[0];
  const int*   ei  = (const int*)d_in[1];
  const int*   src = ei;            // edge_index[0]
  const int*   dst = ei + NE;       // edge_index[1]
  const float* Ws[3]  = {(const float*)d_in[2],  (const float*)d_in[3],  (const float*)d_in[4]};
  const float* bs[3]  = {(const float*)d_in[5],  (const float*)d_in[6],  (const float*)d_in[7]};
  const float* gs[3]  = {(const float*)d_in[8],  (const float*)d_in[9],  (const float*)d_in[10]};
  const float* bts[3] = {(const float*)d_in[11], (const float*)d_in[12], (const float*)d_in[13]};

  char* ws = (char*)d_ws;
  float* dinv = (float*)ws;
  size_t o = ((size_t)NN * 4 + 255) & ~(size_t)255;
  float* xbuf = (float*)(ws + o); o += (size_t)NN * HD * 4;
  float* hbuf = (float*)(ws + o); o += (size_t)NN * HD * 4;
  float* abuf = (float*)(ws + o);

  float* emb_out = (float*)d_out;
  float* x_out   = (float*)d_out + (size_t)NN * HD;

  // degrees -> dinv (in place)
  k_zero<<<(NN + 255) / 256, 256, 0, stream>>>(dinv, NN);
  k_degree<<<(NE + 255) / 256, 256, 0, stream>>>(dst, dinv);
  k_dinv<<<(NN + 255) / 256, 256, 0, stream>>>(dinv);

  const float* xin = x0;
  for (int l = 0; l < 3; ++l) {
    k_gemm<<<NN / 16, 256, 0, stream>>>(xin, Ws[l], hbuf);
    k_zero<<<(NN * HD + 255) / 256, 256, 0, stream>>>(abuf, NN * HD);
    k_scatter<<<((size_t)NE * 32 + 255) / 256, 256, 0, stream>>>(src, dst, dinv, hbuf, abuf);
    const int last = (l == 2);
    k_post<<<(NN * 32 + 255) / 256, 256, 0, stream>>>(
        abuf, hbuf, dinv, bs[l], gs[l], bts[l],
        last ? x_out : xbuf, last ? emb_out : (float*)nullptr, last ? 0 : 1);
    xin = xbuf;
  }
}